// SpikingMaxPool_70720931496129
// MI455X (gfx1250) — compile-verified
//
#include <hip/hip_runtime.h>

// SpikingMaxPool: out[b,ho,wo,c] = max(inputs[sel], 0) where sel is the
// first-argmax of (running_mem+inputs) over the 2x2 window, TF window order.
// Pure streaming kernel: ~462 MB moved -> ~20us floor at 23.3 TB/s.
// CDNA5 path used: GLOBAL_LOAD_ASYNC_TO_LDS_B128 (ASYNCcnt) double-buffered
// staging + non-temporal stores. No WMMA (no matmul structure exists here).

typedef float v4f __attribute__((ext_vector_type(4)));
// The async-LDS builtin's pointer params are GCC-style int vectors:
//   param0: int __attribute__((vector_size(16))) * in global (AS1/"__device__")
//   param1: same pointee in LDS (AS3/"__shared__")
typedef int b128_t __attribute__((vector_size(16)));
typedef __attribute__((address_space(1))) b128_t* glob_b128_ptr;
typedef __attribute__((address_space(3))) b128_t* lds_b128_ptr;

#if __has_builtin(__builtin_amdgcn_global_load_async_to_lds_b128)
#define HAVE_ASYNC_LDS 1
#else
#define HAVE_ASYNC_LDS 0
#endif

#if HAVE_ASYNC_LDS
#define ASYNC_CP16(gsrc, ldst)                                                 \
  __builtin_amdgcn_global_load_async_to_lds_b128(                              \
      (glob_b128_ptr)(gsrc), (lds_b128_ptr)(ldst), 0, 0)
#endif

#if __has_builtin(__builtin_amdgcn_s_wait_asynccnt)
#define WAIT_ASYNC(n) __builtin_amdgcn_s_wait_asynccnt(n)
#elif HAVE_ASYNC_LDS
#define WAIT_ASYNC(n) asm volatile("s_wait_asynccnt %0" ::"i"(n) : "memory")
#else
#define WAIT_ASYNC(n)                                                          \
  do {                                                                         \
  } while (0)
#endif

namespace {
constexpr int kB = 32, kH = 112, kW = 112, kC = 128;
constexpr int kHo = kH / 2, kWo = kW / 2;
constexpr int kNPix = kB * kHo * kWo;  // 100352 output pixels
constexpr int kWavesPerBlock = 8;
constexpr int kBlockDim = kWavesPerBlock * 32;  // 256 threads = 8 wave32
constexpr int kGrid = 1568;  // 12544 waves -> exactly 8 pixels per wave
}  // namespace

// Stage one output pixel's working set (4 window positions x {in, rm},
// 512B each) into this wave's LDS buffer. Lane l covers channels 4l..4l+3.
__device__ __forceinline__ void stage_pixel(const float* __restrict__ in,
                                            const float* __restrict__ rm,
                                            v4f (*__restrict__ buf)[32],
                                            int p, int lane) {
  const int b = p / (kHo * kWo);
  const int r = p - b * (kHo * kWo);
  const int ho = r / kWo;
  const int wo = r - ho * kWo;
  const int base = (((b * kH) + 2 * ho) * kW + 2 * wo) * kC + 4 * lane;
  const float* g = in + base;
  const float* m = rm + base;
#if HAVE_ASYNC_LDS
  ASYNC_CP16(g, &buf[0][lane]);                 // in (h, w)
  ASYNC_CP16(g + kC, &buf[1][lane]);            // in (h, w+1)
  ASYNC_CP16(g + kW * kC, &buf[2][lane]);       // in (h+1, w)
  ASYNC_CP16(g + kW * kC + kC, &buf[3][lane]);  // in (h+1, w+1)
  ASYNC_CP16(m, &buf[4][lane]);                 // rm (h, w)
  ASYNC_CP16(m + kC, &buf[5][lane]);            // rm (h, w+1)
  ASYNC_CP16(m + kW * kC, &buf[6][lane]);       // rm (h+1, w)
  ASYNC_CP16(m + kW * kC + kC, &buf[7][lane]);  // rm (h+1, w+1)
#else
  buf[0][lane] = *(const v4f*)(g);
  buf[1][lane] = *(const v4f*)(g + kC);
  buf[2][lane] = *(const v4f*)(g + kW * kC);
  buf[3][lane] = *(const v4f*)(g + kW * kC + kC);
  buf[4][lane] = *(const v4f*)(m);
  buf[5][lane] = *(const v4f*)(m + kC);
  buf[6][lane] = *(const v4f*)(m + kW * kC);
  buf[7][lane] = *(const v4f*)(m + kW * kC + kC);
#endif
}

__device__ __forceinline__ void compute_pixel(float* __restrict__ out,
                                              const v4f (*__restrict__ buf)[32],
                                              int p, int lane) {
  const v4f i00 = buf[0][lane];
  const v4f i01 = buf[1][lane];
  const v4f i10 = buf[2][lane];
  const v4f i11 = buf[3][lane];
  const v4f m00 = buf[4][lane] + i00;
  const v4f m01 = buf[5][lane] + i01;
  const v4f m10 = buf[6][lane] + i10;
  const v4f m11 = buf[7][lane] + i11;
  v4f o;
#pragma unroll
  for (int k = 0; k < 4; ++k) {
    // First-argmax tie-breaking (strict '>' for later window slots),
    // window order: (0,0),(0,1),(1,0),(1,1) matching TF flattening.
    float best = m00[k];
    float v = i00[k];
    if (m01[k] > best) { best = m01[k]; v = i01[k]; }
    if (m10[k] > best) { best = m10[k]; v = i10[k]; }
    if (m11[k] > best) { best = m11[k]; v = i11[k]; }
    o[k] = fmaxf(v, 0.0f);  // masked second max-pool collapses to this
  }
  __builtin_nontemporal_store(o, (v4f*)(out + (long)p * kC + 4 * lane));
}

__global__ __launch_bounds__(kBlockDim) void spiking_maxpool_kernel(
    const float* __restrict__ in, const float* __restrict__ rm,
    float* __restrict__ out) {
  // Per-wave private double buffer: 2 x 8 segments x 32 lanes x 16B = 8KB.
  __shared__ v4f lds[kWavesPerBlock][2][8][32];
  const int wv = threadIdx.x >> 5;
  const int lane = threadIdx.x & 31;
  int p = blockIdx.x * kWavesPerBlock + wv;
  const int stride = gridDim.x * kWavesPerBlock;
  if (p >= kNPix) return;

  stage_pixel(in, rm, lds[wv][0], p, lane);
  int cur = 0;
  for (; p < kNPix; p += stride) {
    const int pn = p + stride;
    if (pn < kNPix) {
      stage_pixel(in, rm, lds[wv][cur ^ 1], pn, lane);  // prefetch next pixel
      WAIT_ASYNC(8);  // 16 outstanding -> wait until current buffer's 8 done
    } else {
      WAIT_ASYNC(0);
    }
    asm volatile("" ::: "memory");  // keep ds_loads after the async wait
    compute_pixel(out, lds[wv][cur], p, lane);
    cur ^= 1;
  }
}

extern "C" void kernel_launch(void* const* d_in, const int* in_sizes, int n_in,
                              void* d_out, int out_size, void* d_ws,
                              size_t ws_size, hipStream_t stream) {
  (void)in_sizes;
  (void)n_in;
  (void)out_size;
  (void)d_ws;
  (void)ws_size;
  const float* in = (const float*)d_in[0];
  const float* rm = (const float*)d_in[1];
  float* out = (float*)d_out;
  spiking_maxpool_kernel<<<kGrid, kBlockDim, 0, stream>>>(in, rm, out);
}